// Net_85263690760638
// MI455X (gfx1250) — compile-verified
//
#include <hip/hip_runtime.h>
#include <hip/hip_bf16.h>
#include <math.h>

typedef __attribute__((ext_vector_type(2))) float v2f;
typedef __attribute__((ext_vector_type(8))) float v8f;

__device__ __forceinline__ float elu1(float v) {
    return v > 0.0f ? v : (__expf(v) - 1.0f);
}

// ---------------------------------------------------------------------------
// Zero fill (float4 vectorized, grid-stride)
// ---------------------------------------------------------------------------
__global__ void zero_fill(float* __restrict__ p, long n) {
    long i = (long)blockIdx.x * blockDim.x + threadIdx.x;
    long stride = (long)gridDim.x * blockDim.x;
    long n4 = n >> 2;
    for (long k = i; k < n4; k += stride)
        ((float4*)p)[k] = make_float4(0.f, 0.f, 0.f, 0.f);
    if (i == 0)
        for (long k = n4 << 2; k < n; ++k) p[k] = 0.f;
}

// ---------------------------------------------------------------------------
// Edge aggregation: agg[dst] += h[src].  F is 32 or 64; each thread handles a
// float4 quad of one edge so the 4*F-byte source row load is coalesced across
// F/4 consecutive lanes.  Atomics land in L2 (node table is L2-resident).
// ---------------------------------------------------------------------------
__global__ void edge_scatter_add(const float* __restrict__ h,
                                 const int* __restrict__ src,
                                 const int* __restrict__ dst,
                                 float* __restrict__ agg,
                                 int E, int lg_tpe /* log2(F/4) */) {
    const int tpe_mask = (1 << lg_tpe) - 1;
    long total = (long)E << lg_tpe;
    long i = (long)blockIdx.x * blockDim.x + threadIdx.x;
    long stride = (long)gridDim.x * blockDim.x;
    const long F = 4L << lg_tpe;
    for (; i < total; i += stride) {
        int e = (int)(i >> lg_tpe);
        int q = (int)(i & tpe_mask);
        int s = src[e];
        int d = dst[e];
        float4 v = ((const float4*)(h + (long)s * F))[q];
        float* o = agg + (long)d * F + ((long)q << 2);
        atomicAdd(o + 0, v.x);
        atomicAdd(o + 1, v.y);
        atomicAdd(o + 2, v.z);
        atomicAdd(o + 3, v.w);
    }
}

// ---------------------------------------------------------------------------
// Fused (x + agg) @ W + b -> ELU, using V_WMMA_F32_16X16X4_F32.
// One wave32 computes a 16-node x FOUT tile.
//
// A (16x4 fp32) per ISA layout: lanes 0-15 hold M=lane, K={0,1} in v0,v1;
//                                lanes 16-31 hold M=lane-16, K={2,3}.
// B (4x16 fp32) mirrored:        lanes 0-15 hold N=lane, K={0,1};
//                                lanes 16-31 hold N=lane-16, K={2,3}.
// C/D (16x16 fp32, 8 VGPRs):     lane<16: N=lane, VGPR r = row r;
//                                lane>=16: N=lane-16, VGPR r = row r+8.
// ---------------------------------------------------------------------------
template <int FIN, int FOUT>
__global__ void gin_gemm_elu(const float* __restrict__ x,
                             const float* __restrict__ agg,
                             const float* __restrict__ W,   // FIN x FOUT row-major
                             const float* __restrict__ b,   // FOUT
                             float* __restrict__ out,       // N x FOUT
                             int N) {
    const int lane = threadIdx.x & 31;
    const int wave = (int)((blockIdx.x * blockDim.x + threadIdx.x) >> 5);
    const int node0 = wave * 16;
    if (node0 >= N) return;                 // wave-uniform: EXEC stays all-1s

    const int m = lane & 15;                // A-row / B-col / C-col within tile
    const int khalf = lane >> 4;            // 0 -> K{0,1}, 1 -> K{2,3}
    int node = node0 + m;
    if (node >= N) node = N - 1;            // clamp loads (tail tiles)
    const float* xr = x + (long)node * FIN;
    const float* ar = agg + (long)node * FIN;

    // Preload all A fragments for this node row (kept in VGPRs).
    v2f afrag[FIN / 4];
#pragma unroll
    for (int kb = 0; kb < FIN / 4; ++kb) {
        const int k = kb * 4 + khalf * 2;
        afrag[kb].x = xr[k] + ar[k];
        afrag[kb].y = xr[k + 1] + ar[k + 1];
    }

#pragma unroll
    for (int nb = 0; nb < FOUT / 16; ++nb) {
        const float bval = b[nb * 16 + m];  // bias depends only on output col
        v8f acc;
#pragma unroll
        for (int r = 0; r < 8; ++r) acc[r] = bval;

#pragma unroll
        for (int kb = 0; kb < FIN / 4; ++kb) {
            const int k = kb * 4 + khalf * 2;
            v2f bfrag;
            bfrag.x = W[(k + 0) * FOUT + nb * 16 + m];
            bfrag.y = W[(k + 1) * FOUT + nb * 16 + m];
            acc = __builtin_amdgcn_wmma_f32_16x16x4_f32(
                false, afrag[kb], false, bfrag, (short)0, acc, false, false);
        }

        // ELU + store D
#pragma unroll
        for (int r = 0; r < 8; ++r) {
            const int row = r + (khalf << 3);
            if (node0 + row < N)
                out[(long)(node0 + row) * FOUT + nb * 16 + m] = elu1(acc[r]);
        }
    }
}

// ---------------------------------------------------------------------------
// Graph pooling: gsum[batch[i]] += h3[i] (F = 64), counts via float atomics.
// ---------------------------------------------------------------------------
__global__ void pool_sum(const float* __restrict__ h, const int* __restrict__ batch,
                         float* __restrict__ gsum, int N) {
    long total = (long)N * 16;              // 16 float4 quads per node (F=64)
    long i = (long)blockIdx.x * blockDim.x + threadIdx.x;
    long stride = (long)gridDim.x * blockDim.x;
    for (; i < total; i += stride) {
        int node = (int)(i >> 4);
        int q = (int)(i & 15);
        int g = batch[node];
        float4 v = ((const float4*)(h + (long)node * 64))[q];
        float* o = gsum + (long)g * 64 + ((long)q << 2);
        atomicAdd(o + 0, v.x);
        atomicAdd(o + 1, v.y);
        atomicAdd(o + 2, v.z);
        atomicAdd(o + 3, v.w);
    }
}

__global__ void pool_cnt(const int* __restrict__ batch, float* __restrict__ cnt, int N) {
    int i = blockIdx.x * blockDim.x + threadIdx.x;
    if (i < N) atomicAdd(&cnt[batch[i]], 1.0f);
}

// ---------------------------------------------------------------------------
// MLP head: one 64-thread block per graph.  g = gsum/max(cnt,1);
// g -> elu(g@Wf1+bf1) -> elu(@Wf2+bf2) -> @Wf3+bf3 (3.2 MFLOP total).
// ---------------------------------------------------------------------------
__global__ void mlp_head(const float* __restrict__ gsum, const float* __restrict__ cnt,
                         const float* __restrict__ Wf1, const float* __restrict__ bf1,
                         const float* __restrict__ Wf2, const float* __restrict__ bf2,
                         const float* __restrict__ Wf3, const float* __restrict__ bf3,
                         float* __restrict__ outp, int G) {
    __shared__ float s0[64];
    __shared__ float s1[64];
    __shared__ float s2[32];
    const int g = blockIdx.x;
    if (g >= G) return;
    const int t = threadIdx.x;              // blockDim.x == 64

    float c = cnt[g];
    c = c > 1.0f ? c : 1.0f;
    s0[t] = gsum[(long)g * 64 + t] / c;
    __syncthreads();

    float a1 = bf1[t];
#pragma unroll 8
    for (int f = 0; f < 64; ++f) a1 += s0[f] * Wf1[f * 64 + t];
    s1[t] = elu1(a1);
    __syncthreads();

    if (t < 32) {
        float a2 = bf2[t];
#pragma unroll 8
        for (int f = 0; f < 64; ++f) a2 += s1[f] * Wf2[f * 32 + t];
        s2[t] = elu1(a2);
    }
    __syncthreads();

    if (t == 0) {
        float a3 = bf3[0];
#pragma unroll
        for (int f = 0; f < 32; ++f) a3 += s2[f] * Wf3[f];
        outp[g] = a3;
    }
}

// ---------------------------------------------------------------------------
// Launcher
// ---------------------------------------------------------------------------
extern "C" void kernel_launch(void* const* d_in, const int* in_sizes, int n_in,
                              void* d_out, int out_size, void* d_ws, size_t ws_size,
                              hipStream_t stream) {
    const float* x    = (const float*)d_in[0];
    const int*   ei   = (const int*)d_in[1];
    const int*   batch= (const int*)d_in[2];
    const float* W1   = (const float*)d_in[4];
    const float* b1   = (const float*)d_in[5];
    const float* W2   = (const float*)d_in[6];
    const float* b2   = (const float*)d_in[7];
    const float* W3   = (const float*)d_in[8];
    const float* b3   = (const float*)d_in[9];
    const float* Wf1  = (const float*)d_in[10];
    const float* bf1  = (const float*)d_in[11];
    const float* Wf2  = (const float*)d_in[12];
    const float* bf2  = (const float*)d_in[13];
    const float* Wf3  = (const float*)d_in[14];
    const float* bf3  = (const float*)d_in[15];

    const int N = in_sizes[2];       // batch vector length == num nodes
    const int E = in_sizes[1] / 2;   // edge_index is (2, E)
    const int G = out_size;          // one regression value per graph
    const int* src = ei;
    const int* dst = ei + E;

    // Workspace layout (floats)
    float* ws   = (float*)d_ws;
    float* agg  = ws;                        // N*64 (reused per layer)
    float* h1   = agg  + (long)N * 64;       // N*32
    float* h2   = h1   + (long)N * 32;       // N*64
    float* h3   = h2   + (long)N * 64;       // N*64
    float* gsum = h3   + (long)N * 64;       // G*64
    float* cnt  = gsum + (long)G * 64;       // G

    const int ZB = 2048, SB = 4096, TB = 256;
    const int gemm_blocks16 = ((((N + 15) / 16) * 32) + TB - 1) / TB;

    // ---- Layer 1: F 64 -> 32
    zero_fill<<<ZB, TB, 0, stream>>>(agg, (long)N * 64);
    edge_scatter_add<<<SB, TB, 0, stream>>>(x, src, dst, agg, E, /*lg(64/4)=*/4);
    gin_gemm_elu<64, 32><<<gemm_blocks16, TB, 0, stream>>>(x, agg, W1, b1, h1, N);

    // ---- Layer 2: F 32 -> 64
    zero_fill<<<ZB, TB, 0, stream>>>(agg, (long)N * 32);
    edge_scatter_add<<<SB, TB, 0, stream>>>(h1, src, dst, agg, E, /*lg(32/4)=*/3);
    gin_gemm_elu<32, 64><<<gemm_blocks16, TB, 0, stream>>>(h1, agg, W2, b2, h2, N);

    // ---- Layer 3: F 64 -> 64
    zero_fill<<<ZB, TB, 0, stream>>>(agg, (long)N * 64);
    edge_scatter_add<<<SB, TB, 0, stream>>>(h2, src, dst, agg, E, 4);
    gin_gemm_elu<64, 64><<<gemm_blocks16, TB, 0, stream>>>(h2, agg, W3, b3, h3, N);

    // ---- Pooling (gsum and cnt are contiguous -> single zero pass)
    zero_fill<<<64, TB, 0, stream>>>(gsum, (long)G * 64 + G);
    pool_sum<<<2048, TB, 0, stream>>>(h3, batch, gsum, N);
    pool_cnt<<<(N + TB - 1) / TB, TB, 0, stream>>>(batch, cnt, N);

    // ---- MLP head
    mlp_head<<<G, 64, 0, stream>>>(gsum, cnt, Wf1, bf1, Wf2, bf2, Wf3, bf3,
                                   (float*)d_out, G);
}